// LegacyGVAE_83872121356775
// MI455X (gfx1250) — compile-verified
//
#include <hip/hip_runtime.h>

typedef __attribute__((ext_vector_type(2))) float v2f;
typedef __attribute__((ext_vector_type(8))) float v8f;

#define N_NODES 100000
#define N_EDGES 1200000
#define N_GRAPHS 128
#define IN_CH 64
#define HIDDEN 64
#define LATENT 8

// ---------------------------------------------------------------- zero scratch
__global__ void gvae_zero(float* __restrict__ p, long long n) {
  long long i = (long long)blockIdx.x * blockDim.x + threadIdx.x;
  if (i < n) p[i] = 0.0f;
}

// ---------------------------------------------------------------- h = x @ W (WMMA f32 16x16x4)
// One wave per 16x16 output tile. Grid = (N_NODES/16) * (HIDDEN/16) = 6250*4.
__global__ __launch_bounds__(32) void gvae_gemm_h(const float* __restrict__ x,
                                                  const float* __restrict__ W,
                                                  float* __restrict__ h) {
  const int tile = blockIdx.x;
  const int tm   = tile >> 2;      // 16-row tile of nodes
  const int tn   = tile & 3;       // 16-col tile of HIDDEN
  const int lane = threadIdx.x;    // wave32
  const int hl   = lane >> 4;      // 0 = lanes 0-15, 1 = lanes 16-31
  const int l    = lane & 15;

  const int row = tm * 16 + l;     // A-matrix row for this lane
  const int col = tn * 16 + l;     // B/C/D column for this lane

  v8f acc = {};
#pragma unroll
  for (int k0 = 0; k0 < IN_CH; k0 += 4) {
    const int ka = k0 + hl * 2;    // lower lanes: K=k0,k0+1 ; upper: k0+2,k0+3
    v2f a;
    a.x = x[row * IN_CH + ka + 0];
    a.y = x[row * IN_CH + ka + 1];
    v2f bm;
    bm.x = W[(ka + 0) * HIDDEN + col];
    bm.y = W[(ka + 1) * HIDDEN + col];
    acc = __builtin_amdgcn_wmma_f32_16x16x4_f32(
        /*neg_a=*/false, a, /*neg_b=*/false, bm,
        /*c_mod=*/(short)0, acc, /*reuse_a=*/false, /*reuse_b=*/false);
  }
  // C/D layout: VGPR v holds row v (lanes 0-15) and row v+8 (lanes 16-31)
#pragma unroll
  for (int v = 0; v < 8; ++v) {
    const int r = tm * 16 + v + hl * 8;
    h[r * HIDDEN + col] = acc[v];
  }
}

// ---------------------------------------------------------------- in-degree of dst
__global__ void gvae_degree(const int* __restrict__ ei, float* __restrict__ deg, int E) {
  int e = blockIdx.x * blockDim.x + threadIdx.x;
  if (e < E) atomicAdd(&deg[ei[E + e]], 1.0f);  // row 1 = dst
}

// ---------------------------------------------------------------- scatter messages
// one thread per (edge, channel); deg is tiny (400 KB) so rsqrt inputs hit cache
__global__ void gvae_scatter(const int* __restrict__ ei, const float* __restrict__ h,
                             const float* __restrict__ deg, float* __restrict__ agg, int E) {
  long long tid = (long long)blockIdx.x * blockDim.x + threadIdx.x;
  if (tid >= (long long)E * HIDDEN) return;
  const int e = (int)(tid >> 6);
  const int c = (int)(tid & 63);
  const int s = ei[e];
  const int d = ei[E + e];
  const float norm = rsqrtf(deg[s] + 1.0f) * rsqrtf(deg[d] + 1.0f);
  atomicAdd(&agg[(long long)d * HIDDEN + c], h[(long long)s * HIDDEN + c] * norm);
}

// ---------------------------------------------------------------- relu(agg + h/deg + b) -> pooled sums
__global__ void gvae_relu_pool(const float* __restrict__ agg, const float* __restrict__ h,
                               const float* __restrict__ deg, const float* __restrict__ b,
                               const int* __restrict__ batch,
                               float* __restrict__ pool, float* __restrict__ counts, int N) {
  long long tid = (long long)blockIdx.x * blockDim.x + threadIdx.x;
  if (tid >= (long long)N * HIDDEN) return;
  const int n = (int)(tid >> 6);
  const int c = (int)(tid & 63);
  const float dg = deg[n] + 1.0f;                      // self-loop degree
  float v = agg[(long long)n * HIDDEN + c] + h[(long long)n * HIDDEN + c] / dg + b[c];
  v = fmaxf(v, 0.0f);
  const int g = batch[n];
  atomicAdd(&pool[g * HIDDEN + c], v);
  if (c == 0) atomicAdd(&counts[g], 1.0f);
}

// ---------------------------------------------------------------- VAE heads + decode (1 thread / graph)
__global__ void gvae_finalize(const float* __restrict__ pool, const float* __restrict__ counts,
                              const float* __restrict__ eps,
                              const float* __restrict__ mu_w, const float* __restrict__ mu_b,
                              const float* __restrict__ lv_w, const float* __restrict__ lv_b,
                              const float* __restrict__ dec_w, const float* __restrict__ dec_b,
                              float* __restrict__ out, int G) {
  const int g = blockIdx.x * blockDim.x + threadIdx.x;
  if (g >= G) return;
  const float cnt = fmaxf(counts[g], 1.0f);
  float p[HIDDEN];
#pragma unroll
  for (int c = 0; c < HIDDEN; ++c) p[c] = pool[g * HIDDEN + c] / cnt;

  float mu[LATENT], lv[LATENT], z[LATENT];
#pragma unroll
  for (int j = 0; j < LATENT; ++j) {
    float m = mu_b[j], l = lv_b[j];
    for (int c = 0; c < HIDDEN; ++c) {
      m = fmaf(p[c], mu_w[c * LATENT + j], m);
      l = fmaf(p[c], lv_w[c * LATENT + j], l);
    }
    mu[j] = m; lv[j] = l;
    z[j] = m + eps[g * LATENT + j] * __expf(0.5f * l);
  }

  float* adj = out;                       // G*100
  float* muo = out + G * 100;             // G*8
  float* lvo = out + G * 100 + G * LATENT;
  for (int k = 0; k < 100; ++k) {
    float lg = dec_b[k];
#pragma unroll
    for (int j = 0; j < LATENT; ++j) lg = fmaf(z[j], dec_w[j * 100 + k], lg);
    adj[g * 100 + k] = (lg > 0.0f) ? 1.0f : 0.0f;
  }
#pragma unroll
  for (int j = 0; j < LATENT; ++j) { muo[g * LATENT + j] = mu[j]; lvo[g * LATENT + j] = lv[j]; }
}

// ---------------------------------------------------------------- launch
extern "C" void kernel_launch(void* const* d_in, const int* in_sizes, int n_in,
                              void* d_out, int out_size, void* d_ws, size_t ws_size,
                              hipStream_t stream) {
  const float* x     = (const float*)d_in[0];
  const int*   ei    = (const int*)d_in[1];   // [2, E]
  const int*   batch = (const int*)d_in[2];
  const float* eps   = (const float*)d_in[3];
  const float* W     = (const float*)d_in[4];
  const float* b     = (const float*)d_in[5];
  const float* mu_w  = (const float*)d_in[6];
  const float* mu_b  = (const float*)d_in[7];
  const float* lv_w  = (const float*)d_in[8];
  const float* lv_b  = (const float*)d_in[9];
  const float* dec_w = (const float*)d_in[10];
  const float* dec_b = (const float*)d_in[11];
  float* out = (float*)d_out;

  // workspace layout (floats)
  float* ws      = (float*)d_ws;
  float* h       = ws;                                   // N*64
  float* agg     = h + (long long)N_NODES * HIDDEN;      // N*64
  float* deg     = agg + (long long)N_NODES * HIDDEN;    // N
  float* pool    = deg + N_NODES;                        // G*64
  float* counts  = pool + N_GRAPHS * HIDDEN;             // G

  // zero agg|deg|pool|counts (contiguous)
  const long long nz = (long long)N_NODES * HIDDEN + N_NODES + N_GRAPHS * HIDDEN + N_GRAPHS;
  gvae_zero<<<(int)((nz + 255) / 256), 256, 0, stream>>>(agg, nz);

  // h = x @ W via f32 WMMA
  gvae_gemm_h<<<(N_NODES / 16) * (HIDDEN / 16), 32, 0, stream>>>(x, W, h);

  // degrees
  gvae_degree<<<(N_EDGES + 255) / 256, 256, 0, stream>>>(ei, deg, N_EDGES);

  // edge messages -> agg
  const long long et = (long long)N_EDGES * HIDDEN;
  gvae_scatter<<<(int)((et + 255) / 256), 256, 0, stream>>>(ei, h, deg, agg, N_EDGES);

  // relu + self-loop + bias, pooled sums
  const long long nt = (long long)N_NODES * HIDDEN;
  gvae_relu_pool<<<(int)((nt + 255) / 256), 256, 0, stream>>>(agg, h, deg, b, batch,
                                                              pool, counts, N_NODES);

  // heads + decode
  gvae_finalize<<<1, 128, 0, stream>>>(pool, counts, eps, mu_w, mu_b, lv_w, lv_b,
                                       dec_w, dec_b, out, N_GRAPHS);
}